// VariableMemoryProjection_52956946760025
// MI455X (gfx1250) — compile-verified
//
#include <hip/hip_runtime.h>
#include <hip/hip_bf16.h>
#include <stddef.h>

// ---------------------------------------------------------------------------
// HiPPO-LegS bilinear SSM scan, CDNA5 (gfx1250, wave32).
//
// Structure exploited: A = -tril(r r^T, -1) - diag(n+1), r_n = sqrt(2n+1).
// (I - h A) is lower-triangular + rank-1 strictly-lower ==> forward
// substitution is an O(N) scalar recurrence per column, and
//   As = (I-hA)^{-1}(I+hA) = 2 (I-hA)^{-1} - I
//   u  = in*dt*(I-hA)^{-1} r
// Blocked parallel scan over L: chunk transition matrices P_g accumulated
// with V_WMMA_F32_16X16X4_F32 (fp32 matches reference precision).
// ---------------------------------------------------------------------------

#define LSEQ 1024
#define BM   64      // B*M = 4*16 chains
#define NN   64      // HiPPO order
#define CHK  128     // chunk length
#define NG   8       // LSEQ / CHK

typedef float v2f __attribute__((ext_vector_type(2)));
typedef float v8f __attribute__((ext_vector_type(8)));

// Build column t of As = 2*(I-hA)^{-1} - I into LDS (stride-65 padded).
// Exact forward substitution on the rank-1 semiseparable lower factor.
__device__ __forceinline__ void build_As_col(float* sAs, const float* rs,
                                             const float* invd, float h, int t) {
  float q = 0.0f;
  #pragma unroll 4
  for (int n = 0; n < NN; ++n) {
    float o;
    if (n < t) {
      o = 0.0f;
    } else {
      float e = (n == t) ? 1.0f : 0.0f;
      float val = (e - h * rs[n] * q) * invd[n];   // col t of (I-hA)^{-1}
      q += rs[n] * val;
      o = 2.0f * val - e;                           // As = 2*inv - I
    }
    sAs[n * 65 + t] = o;
  }
}

// ---------------------------------------------------------------------------
// Phase 1: per (chain, chunk): local zero-init state z over the chunk AND the
// chunk transition matrix P = As_last ... As_first via WMMA fp32 matmuls.
// Block = 64 threads = 2 waves; each wave owns 8 of the 16 output tiles.
// ---------------------------------------------------------------------------
__global__ __launch_bounds__(64) void hippo_phase1(const float* __restrict__ xin,
                                                   const float* __restrict__ dtin,
                                                   float* __restrict__ Pws,
                                                   float* __restrict__ Zws) {
  __shared__ float sAs[NN * 65];
  __shared__ float sP[2][NN * 65];
  __shared__ float invd[NN], wv[NN], sz[NN], rs[NN];

  const int t     = threadIdx.x;
  const int chain = blockIdx.x >> 3;     // /NG
  const int g     = blockIdx.x & (NG - 1);
  const int lane  = t & 31;
  const int l15   = lane & 15;
  const int hi    = lane >> 4;           // lane half: 0 or 1
  const int wave  = t >> 5;

  rs[t] = sqrtf(2.0f * (float)t + 1.0f);
  sz[t] = 0.0f;
  for (int k = 0; k < NN; ++k) sP[0][t * 65 + k] = (k == t) ? 1.0f : 0.0f;
  __syncthreads();

  int cur = 0;
  for (int step = 0; step < CHK; ++step) {
    const int   l   = g * CHK + step;
    const float dtv = dtin[l * BM + chain];
    const float uin = xin[l * BM + chain];
    const float h   = 0.5f * dtv;
    const float c   = 0.5f * uin * dtv;  // u = c*(As r + r)

    invd[t] = 1.0f / (1.0f + h * (float)(t + 1));
    wv[t]   = sz[t] + c * rs[t];
    __syncthreads();

    build_As_col(sAs, rs, invd, h, t);
    __syncthreads();

    // z_new[t] = sum_k As[t,k]*(z[k] + c r[k]) + c r[t]
    float znew = c * rs[t];
    #pragma unroll 8
    for (int k = 0; k < NN; ++k) znew += sAs[t * 65 + k] * wv[k];

    // P_next = As @ P_cur  (64x64x64 fp32, WMMA 16x16x4, K in steps of 4)
    {
      const float* Pc = sP[cur];
      float*       Pn = sP[cur ^ 1];
      for (int tt = 0; tt < 8; ++tt) {
        const int tile = wave * 8 + tt;
        const int ti = tile >> 2, tj = tile & 3;
        v8f acc = {};
        #pragma unroll 4
        for (int kc = 0; kc < 16; ++kc) {
          // A frag 16x4: lanes 0-15 K={0,1}, lanes 16-31 K={2,3}
          const int ar = ti * 16 + l15;
          const int ak = kc * 4 + hi * 2;
          v2f a; a.x = sAs[ar * 65 + ak]; a.y = sAs[ar * 65 + ak + 1];
          // B frag 4x16: VGPR i -> row K=i (lanes 0-15), K=i+2 (lanes 16-31)
          const int bc = tj * 16 + l15;
          v2f b; b.x = Pc[(kc * 4 + hi * 2 + 0) * 65 + bc];
          b.y       = Pc[(kc * 4 + hi * 2 + 1) * 65 + bc];
          acc = __builtin_amdgcn_wmma_f32_16x16x4_f32(
              false, a, false, b, (short)0, acc, false, false);
        }
        #pragma unroll
        for (int i = 0; i < 8; ++i)
          Pn[(ti * 16 + hi * 8 + i) * 65 + tj * 16 + l15] = acc[i];
      }
    }
    sz[t] = znew;
    __syncthreads();
    cur ^= 1;
  }

  float* Pout = Pws + (size_t)(chain * NG + g) * (NN * NN);
  for (int k = 0; k < NN; ++k) Pout[t * NN + k] = sP[cur][t * 65 + k];
  Zws[(chain * NG + g) * NN + t] = sz[t];
}

// ---------------------------------------------------------------------------
// Phase 2: per chain, chain carries across the NG chunks:
//   carry_in[0]=0;  carry_in[g+1] = P_g @ carry_in[g] + zend_g
// ---------------------------------------------------------------------------
__global__ __launch_bounds__(64) void hippo_phase2(const float* __restrict__ Pws,
                                                   const float* __restrict__ Zws,
                                                   float* __restrict__ Cws) {
  __shared__ float carry[NN];
  const int t = threadIdx.x;
  const int chain = blockIdx.x;
  carry[t] = 0.0f;
  __syncthreads();
  for (int g = 0; g < NG; ++g) {
    Cws[(chain * NG + g) * NN + t] = carry[t];
    const float* P = Pws + (size_t)(chain * NG + g) * (NN * NN);
    float dot = Zws[(chain * NG + g) * NN + t];
    #pragma unroll 8
    for (int k = 0; k < NN; ++k) dot += P[t * NN + k] * carry[k];
    __syncthreads();
    carry[t] = dot;
    __syncthreads();
  }
}

// ---------------------------------------------------------------------------
// Phase 3: per (chain, chunk): replay the chunk from its true carry and
// stream states to the output.
// ---------------------------------------------------------------------------
__global__ __launch_bounds__(64) void hippo_phase3(const float* __restrict__ xin,
                                                   const float* __restrict__ dtin,
                                                   const float* __restrict__ Cws,
                                                   float* __restrict__ out) {
  __shared__ float sAs[NN * 65];
  __shared__ float invd[NN], wv[NN], sz[NN], rs[NN];
  const int t     = threadIdx.x;
  const int chain = blockIdx.x >> 3;
  const int g     = blockIdx.x & (NG - 1);

  rs[t] = sqrtf(2.0f * (float)t + 1.0f);
  sz[t] = Cws[(chain * NG + g) * NN + t];
  __syncthreads();

  for (int step = 0; step < CHK; ++step) {
    const int   l   = g * CHK + step;
    const float dtv = dtin[l * BM + chain];
    const float uin = xin[l * BM + chain];
    const float h   = 0.5f * dtv;
    const float c   = 0.5f * uin * dtv;

    invd[t] = 1.0f / (1.0f + h * (float)(t + 1));
    wv[t]   = sz[t] + c * rs[t];
    __syncthreads();

    build_As_col(sAs, rs, invd, h, t);
    __syncthreads();

    float snew = c * rs[t];
    #pragma unroll 8
    for (int k = 0; k < NN; ++k) snew += sAs[t * 65 + k] * wv[k];

    out[(size_t)(l * BM + chain) * NN + t] = snew;
    sz[t] = snew;
    __syncthreads();
  }
}

// ---------------------------------------------------------------------------
// Fallback (if workspace too small): fully sequential per-chain recurrence.
// ---------------------------------------------------------------------------
__global__ __launch_bounds__(64) void hippo_seq(const float* __restrict__ xin,
                                                const float* __restrict__ dtin,
                                                float* __restrict__ out) {
  __shared__ float sAs[NN * 65];
  __shared__ float invd[NN], wv[NN], sz[NN], rs[NN];
  const int t = threadIdx.x;
  const int chain = blockIdx.x;

  rs[t] = sqrtf(2.0f * (float)t + 1.0f);
  sz[t] = 0.0f;
  __syncthreads();

  for (int l = 0; l < LSEQ; ++l) {
    const float dtv = dtin[l * BM + chain];
    const float uin = xin[l * BM + chain];
    const float h   = 0.5f * dtv;
    const float c   = 0.5f * uin * dtv;
    invd[t] = 1.0f / (1.0f + h * (float)(t + 1));
    wv[t]   = sz[t] + c * rs[t];
    __syncthreads();
    build_As_col(sAs, rs, invd, h, t);
    __syncthreads();
    float snew = c * rs[t];
    #pragma unroll 8
    for (int k = 0; k < NN; ++k) snew += sAs[t * 65 + k] * wv[k];
    out[(size_t)(l * BM + chain) * NN + t] = snew;
    sz[t] = snew;
    __syncthreads();
  }
}

extern "C" void kernel_launch(void* const* d_in, const int* in_sizes, int n_in,
                              void* d_out, int out_size, void* d_ws, size_t ws_size,
                              hipStream_t stream) {
  const float* xin  = (const float*)d_in[0];  // inputs (L,B,M)
  const float* dtin = (const float*)d_in[1];  // dt     (L,B,M)
  float* out = (float*)d_out;                 // (L,B,M,N)

  const size_t pElems = (size_t)BM * NG * NN * NN;  // chunk transition matrices
  const size_t vElems = (size_t)BM * NG * NN;       // zend / carries
  const size_t needBytes = (pElems + 2 * vElems) * sizeof(float);

  if (ws_size >= needBytes) {
    float* Pws = (float*)d_ws;
    float* Zws = Pws + pElems;
    float* Cws = Zws + vElems;
    hippo_phase1<<<BM * NG, 64, 0, stream>>>(xin, dtin, Pws, Zws);
    hippo_phase2<<<BM, 64, 0, stream>>>(Pws, Zws, Cws);
    hippo_phase3<<<BM * NG, 64, 0, stream>>>(xin, dtin, Cws, out);
  } else {
    hippo_seq<<<BM, 64, 0, stream>>>(xin, dtin, out);
  }
}